// APE_61555471286335
// MI455X (gfx1250) — compile-verified
//
#include <hip/hip_runtime.h>

// Problem constants (match reference)
#define B_SZ   16384
#define NNEG   10
#define D_DIM  8
#define E_DIM  64
#define NSAMP  (B_SZ * (1 + NNEG))   // 180224 samples total
#define NTILE  (NSAMP / 2)           // 90112 waves, 2 samples per wave

typedef __attribute__((ext_vector_type(2))) float v2f;
typedef __attribute__((ext_vector_type(8))) float v8f;

// One wave32 per tile of 2 samples.
// A tile: 16 rows (2 samples x 8 dims) x 64 cols (embedding dim), fp32.
// gram = A * A^T accumulated with 16 chained v_wmma_f32_16x16x4_f32.
__global__ __launch_bounds__(256) void APE_score_wmma_kernel(
    const int*   __restrict__ pos_x,      // [B, D]
    const int*   __restrict__ neg_x,      // [B, NNEG, D]
    const float* __restrict__ emb,        // [N_ENT, E]
    const float* __restrict__ pair_w,     // [N_PAIRS]
    const float* __restrict__ c_scalar,   // [1]
    float*       __restrict__ out)        // [NSAMP] = pos ++ neg (flat)
{
    const int tid  = blockIdx.x * blockDim.x + threadIdx.x;
    const int lane = threadIdx.x & 31;
    const int tile = tid >> 5;
    if (tile >= NTILE) return;            // never taken (exact grid); EXEC stays full

    // Which tile-row this lane feeds, and which K-pair (A/B lane-half K split).
    const int row  = lane & 15;           // 0..15 within the 16-row tile
    const int koff = (lane >> 4) * 2;     // lanes 0-15 -> k%4 in {0,1}; 16-31 -> {2,3}
    const int s    = tile * 2 + (row >> 3);   // global sample id
    const int d    = row & 7;             // dimension index within sample

    // Branchless pointer select (v_cndmask): both samples of a tile are on the
    // same side of B (B is even), so this is uniform anyway.
    const int* xp = (s < B_SZ) ? (pos_x + s * D_DIM + d)
                               : (neg_x + (size_t)(s - B_SZ) * D_DIM + d);
    const int idx = *xp;
    const float* rowp = emb + (long long)idx * E_DIM + koff;

    // Load this lane's 32 floats (16 aligned float2 chunks, global_load_b64).
    v2f a[16];
#pragma unroll
    for (int t = 0; t < 16; ++t)
        a[t] = *reinterpret_cast<const v2f*>(rowp + 4 * t);

    // gram += A_slice * A_slice^T, K advancing 4 per step.
    // For this op the A and B VGPR layouts coincide when B == A^T,
    // so the same registers serve as both operands.
    v8f acc = {};
#pragma unroll
    for (int t = 0; t < 16; ++t)
        acc = __builtin_amdgcn_wmma_f32_16x16x4_f32(
                  /*neg_a=*/false, a[t],
                  /*neg_b=*/false, a[t],
                  /*c_mod=*/(short)0, acc,
                  /*reuse_a=*/false, /*reuse_b=*/false);

    // C layout: lane<16 -> acc[r] = gram[r][lane]; lane>=16 -> gram[r+8][lane-16].
    // Sample 0 block (rows 0-7, cols 0-7): lanes 0-7, block col j = lane.
    // Sample 1 block (rows 8-15, cols 8-15): lanes 24-31, block col j = lane-24.
    const int j = (lane < 8) ? lane : ((lane >= 24) ? (lane - 24) : 0);
    float partial = 0.f;
#pragma unroll
    for (int r = 0; r < 7; ++r)
        partial += (r < j) ? acc[r] : 0.f;

    // Butterfly-reduce within aligned groups of 8 lanes.
    partial += __shfl_xor(partial, 1, 32);
    partial += __shfl_xor(partial, 2, 32);
    partial += __shfl_xor(partial, 4, 32);

    const float w  = __expf(pair_w[0]);    // scalar, L2/K$ resident
    const float cc = c_scalar[0];
    if (lane == 0)
        out[tile * 2 + 0] = __expf(w * partial + cc);
    else if (lane == 24)
        out[tile * 2 + 1] = __expf(w * partial + cc);
}

extern "C" void kernel_launch(void* const* d_in, const int* in_sizes, int n_in,
                              void* d_out, int out_size, void* d_ws, size_t ws_size,
                              hipStream_t stream) {
    const int*   pos_x = (const int*)  d_in[0];   // [B, D] int32
    const int*   neg_x = (const int*)  d_in[1];   // [B, NNEG, D] int32
    const float* emb   = (const float*)d_in[2];   // [N_ENT, E] f32
    const float* pw    = (const float*)d_in[3];   // [N_PAIRS] f32
    const float* cs    = (const float*)d_in[4];   // [1] f32
    float* out = (float*)d_out;                   // [NSAMP] f32

    const int waves_per_block = 256 / 32;                 // 8 waves (wave32)
    const int blocks = NTILE / waves_per_block;           // 90112/8 = 11264 exact
    APE_score_wmma_kernel<<<blocks, 256, 0, stream>>>(pos_x, neg_x, emb, pw, cs, out);
}